// Ensemble_highway_14577119003463
// MI455X (gfx1250) — compile-verified
//
#include <hip/hip_runtime.h>
#include <hip/hip_bf16.h>
#include <math.h>

// Problem constants (from the reference)
#define NC 8      // clients
#define BB 8192   // batch
#define CC 1000   // classes
#define FF 2048   // features

typedef float v2f __attribute__((ext_vector_type(2)));
typedef float v8f __attribute__((ext_vector_type(8)));

__device__ __forceinline__ float sigmoidf(float x) {
    return 1.0f / (1.0f + __expf(-x));
}

// ---------------------------------------------------------------------------
// Kernel 1: w[n][b] = sigmoid( dot(clients_logit[n][b][:], ensemble_scale) + eb )
// One wave handles 16 consecutive b-rows of one client using
// V_WMMA_F32_16X16X4_F32 to do the K-reduction over C.
//   A (16x4, MxK): lanes 0-15 -> M=lane, K=0,1 in the 2 VGPRs;
//                  lanes 16-31 -> M=lane-16, K=2,3.  => float2 load at c + 2*(lane>>4)
//   B (4x16, KxN): broadcast scale over all N columns with the same koff trick.
//   D (16x16): every column equal; lane 0 holds rows 0..7 in acc[0..7],
//              lane 16 holds rows 8..15.
// ---------------------------------------------------------------------------
__global__ __launch_bounds__(256) void w_kernel(
    const float* __restrict__ clogit,   // [NC][BB][CC]
    const float* __restrict__ escale,   // [CC] (shape [C,1] flat)
    const float* __restrict__ ebias,    // [1]
    float* __restrict__ w)              // [NC][BB]
{
    const int lane   = threadIdx.x & 31;
    const int waveId = blockIdx.x * 8 + (threadIdx.x >> 5);
    const int n      = waveId >> 9;          // 512 tiles per client
    const int tile   = waveId & 511;
    const int row    = tile * 16 + (lane & 15);
    const int koff   = (lane >> 4) * 2;      // 0 for lanes 0-15, 2 for lanes 16-31
    const float* arow = clogit + (long long)(n * BB + row) * CC + koff;

    v8f acc = {};
    for (int c = 0; c < CC; c += 4) {
        v2f a = *(const v2f*)(arow + c);                 // 8B aligned (koff even)
        v2f b = *(const v2f*)(escale + c + koff);        // broadcast over N
        __builtin_prefetch(arow + c + 256, 0, 1);        // ~64 iters ahead
        acc = __builtin_amdgcn_wmma_f32_16x16x4_f32(
            false, a, false, b, (short)0, acc, false, false);
    }

    const float eb = ebias[0];
    if ((lane & 15) == 0) {                  // lanes 0 and 16
        const int rbase = tile * 16 + (lane >> 4) * 8;
        float* wp = w + (long long)n * BB + rbase;
#pragma unroll
        for (int r = 0; r < 8; ++r)
            wp[r] = sigmoidf(acc[r] + eb);
    }
}

// ---------------------------------------------------------------------------
// Kernel 2: carry[b] = sigmoid( (1/8) * sum_n dot(feature[n][b][:], W2) + b2 )
// Block = 8 waves; wave i reduces client i over F with WMMA, then an LDS
// reduction combines clients.
// ---------------------------------------------------------------------------
__global__ __launch_bounds__(256) void carry_kernel(
    const float* __restrict__ cfeat,    // [NC][BB][FF]
    const float* __restrict__ W2,       // [FF] (shape [F,1] flat)
    const float* __restrict__ b2,       // [1]
    float* __restrict__ carry)          // [BB]
{
    __shared__ float red[NC][16];
    const int lane = threadIdx.x & 31;
    const int n    = threadIdx.x >> 5;       // client = wave id
    const int tile = blockIdx.x;             // 512 tiles of 16 rows
    const int row  = tile * 16 + (lane & 15);
    const int koff = (lane >> 4) * 2;
    const float* arow = cfeat + (long long)(n * BB + row) * FF + koff;

    v8f acc = {};
    for (int f = 0; f < FF; f += 4) {
        v2f a = *(const v2f*)(arow + f);
        v2f b = *(const v2f*)(W2 + f + koff);
        __builtin_prefetch(arow + f + 512, 0, 1);
        acc = __builtin_amdgcn_wmma_f32_16x16x4_f32(
            false, a, false, b, (short)0, acc, false, false);
    }

    if ((lane & 15) == 0) {
        const int rb = (lane >> 4) * 8;
#pragma unroll
        for (int r = 0; r < 8; ++r) red[n][rb + r] = acc[r];
    }
    __syncthreads();

    if (threadIdx.x < 16) {
        float s = 0.0f;
#pragma unroll
        for (int nn = 0; nn < NC; ++nn) s += red[nn][threadIdx.x];
        carry[tile * 16 + threadIdx.x] = sigmoidf(s * 0.125f + b2[0]);
    }
}

// ---------------------------------------------------------------------------
// Kernel 3: fused output. One block per b-row; threads 0..249 each handle a
// float4 chunk of C. Inline L1-normalization of w (sigmoid>0 => |w|=w).
// out = carry*(Σ_n w̄·cl)*ls + lb) + (1-carry)*new_logit
// ---------------------------------------------------------------------------
__global__ __launch_bounds__(256) void fuse_kernel(
    const float* __restrict__ clogit,    // [NC][BB][CC]
    const float* __restrict__ nlogit,    // [BB][CC]
    const float* __restrict__ lscale,    // [CC]
    const float* __restrict__ lbias,     // [CC]
    const float* __restrict__ w,         // [NC][BB]
    const float* __restrict__ carry,     // [BB]
    float* __restrict__ out)             // [BB][CC]
{
    const int b = blockIdx.x;
    const int t = threadIdx.x;

    float wn[NC];
    float wsum = 0.0f;
#pragma unroll
    for (int n = 0; n < NC; ++n) { wn[n] = w[(long long)n * BB + b]; wsum += wn[n]; }
    const float inv = 1.0f / wsum;
    const float cr  = carry[b];

    if (t < CC / 4) {
        const int c = t * 4;
        float ax = 0.f, ay = 0.f, az = 0.f, aw = 0.f;
#pragma unroll
        for (int n = 0; n < NC; ++n) {
            const float4 cl = *(const float4*)(clogit + (long long)(n * BB + b) * CC + c);
            const float wv = wn[n] * inv;
            ax = fmaf(wv, cl.x, ax);
            ay = fmaf(wv, cl.y, ay);
            az = fmaf(wv, cl.z, az);
            aw = fmaf(wv, cl.w, aw);
        }
        const float4 ls = *(const float4*)(lscale + c);
        const float4 lb = *(const float4*)(lbias + c);
        const float4 nl = *(const float4*)(nlogit + (long long)b * CC + c);
        const float  omc = 1.0f - cr;
        float4 o;
        o.x = fmaf(cr, fmaf(ax, ls.x, lb.x), omc * nl.x);
        o.y = fmaf(cr, fmaf(ay, ls.y, lb.y), omc * nl.y);
        o.z = fmaf(cr, fmaf(az, ls.z, lb.z), omc * nl.z);
        o.w = fmaf(cr, fmaf(aw, ls.w, lb.w), omc * nl.w);
        *(float4*)(out + (long long)b * CC + c) = o;
    }
}

extern "C" void kernel_launch(void* const* d_in, const int* in_sizes, int n_in,
                              void* d_out, int out_size, void* d_ws, size_t ws_size,
                              hipStream_t stream) {
    const float* cfeat  = (const float*)d_in[0];  // [8,8192,2048]
    const float* clogit = (const float*)d_in[1];  // [8,8192,1000]
    const float* nlogit = (const float*)d_in[2];  // [8192,1000]
    const float* escale = (const float*)d_in[3];  // [1000,1]
    const float* ebias  = (const float*)d_in[4];  // [1]
    const float* lscale = (const float*)d_in[5];  // [1000]
    const float* lbias  = (const float*)d_in[6];  // [1000]
    const float* W2     = (const float*)d_in[7];  // [2048,1]
    const float* b2     = (const float*)d_in[8];  // [1]
    float* out = (float*)d_out;

    float* w     = (float*)d_ws;          // [8][8192]  = 256 KB
    float* carry = w + NC * BB;           // [8192]     =  32 KB

    // 8*512 = 4096 waves total, 8 waves per block
    w_kernel<<<512, 256, 0, stream>>>(clogit, escale, ebias, w);
    // 512 blocks x 8 waves (one client per wave), LDS reduce
    carry_kernel<<<512, 256, 0, stream>>>(cfeat, W2, b2, carry);
    // one block per batch row
    fuse_kernel<<<BB, 256, 0, stream>>>(clogit, nlogit, lscale, lbias, w, carry, out);
}